// MambaBlock_80874234183824
// MI455X (gfx1250) — compile-verified
//
#include <hip/hip_runtime.h>
#include <math.h>

// ---------------- Problem constants ----------------
#define DIMV     256
#define D_STATE  16
#define D_CONV   4
#define D_INNER  512
#define DT_RANK  16
#define B_SZ     4
#define L_SEQ    2048
#define MROWS    (B_SZ * L_SEQ)      // 8192

typedef __attribute__((ext_vector_type(2))) float v2f;
typedef __attribute__((ext_vector_type(8))) float v8f;

// ---------------- WMMA f32 16x16x4 multi-tile GEMM core ----------------
// Each wave computes a 16(M) x (NT*16)(N) block:
//   C[m0:m0+16, n0:n0+NT*16] = A[m0:m0+16, :K] * W[n0:n0+NT*16, :K]^T
// A row-major (lda); W row-major (N x K) so B[k][n] = W[n*K+k].
//
// Fragment layouts (ISA 7.12.2, 32-bit A 16x4):
//   A: lanes 0-15 -> M=lane,   K=k0+0,k0+1 ; lanes 16-31 -> M=lane-16, K=k0+2,k0+3
//   B mirrors (rows striped across lanes).
//   C/D: VGPR r: lanes 0-15 -> M=r, N=lane ; lanes 16-31 -> M=r+8, N=lane-16.
template<int KDIM, int NT>
__device__ inline void wmma_tiles_f32(const float* __restrict__ A, int lda,
                                      const float* __restrict__ W,
                                      int m0, int n0, v8f acc[NT]) {
    const int lane = threadIdx.x & 31;
    const int lm   = lane & 15;
    const int koff = (lane >> 4) << 1;          // 0 for lanes 0-15, 2 for 16-31
    const float* arow  = A + (size_t)(m0 + lm) * (size_t)lda + koff;
    const float* wbase = W + (size_t)(n0 + lm) * (size_t)KDIM + koff;

    if constexpr (KDIM <= 32) {
#pragma unroll
        for (int k = 0; k < KDIM; k += 4) {
            v2f a = *(const v2f*)(arow + k);
#pragma unroll
            for (int t = 0; t < NT; ++t) {
                v2f b = *(const v2f*)(wbase + (size_t)t * (16 * KDIM) + k);
                acc[t] = __builtin_amdgcn_wmma_f32_16x16x4_f32(
                             false, a, false, b, (short)0, acc[t], false, false);
            }
        }
    } else {
        for (int kk = 0; kk < KDIM; kk += 32) {
            // Speculative prefetch of the next K-chunk (global_prefetch_b8).
            __builtin_prefetch(arow + kk + 32, 0, 1);
            __builtin_prefetch(wbase + kk + 32, 0, 1);
#pragma unroll
            for (int k4 = 0; k4 < 32; k4 += 4) {
                const int k = kk + k4;
                v2f a = *(const v2f*)(arow + k);
#pragma unroll
                for (int t = 0; t < NT; ++t) {
                    v2f b = *(const v2f*)(wbase + (size_t)t * (16 * KDIM) + k);
                    acc[t] = __builtin_amdgcn_wmma_f32_16x16x4_f32(
                                 false, a, false, b, (short)0, acc[t], false, false);
                }
            }
        }
    }
}

__device__ inline void store_tile_f32(float* __restrict__ C, int ldc,
                                      int m0, int n0, v8f acc) {
    const int lane = threadIdx.x & 31;
    const int lm   = lane & 15;
    const int rofs = (lane >> 4) << 3;          // 0 or 8
#pragma unroll
    for (int r = 0; r < 8; ++r)
        C[(size_t)(m0 + r + rofs) * (size_t)ldc + (n0 + lm)] = acc[r];
}

// ---------------- Generic WMMA GEMM kernel ----------------
// block = 128 threads = 4 waves; wave w -> M-tile (blockIdx.x*4 + w),
// each wave covers NT consecutive N-tiles (A-fragment reuse, 1+NT loads
// per NT WMMAs). grid.x = M/64, grid.y = N/(NT*16).
template<int KDIM, int NT>
__global__ void gemm_wmma_kernel(const float* __restrict__ A, int lda,
                                 const float* __restrict__ W,
                                 float* __restrict__ C, int ldc) {
    const int m0 = (blockIdx.x * 4 + (threadIdx.x >> 5)) * 16;
    const int n0 = blockIdx.y * (NT * 16);
    v8f acc[NT];
#pragma unroll
    for (int t = 0; t < NT; ++t) acc[t] = (v8f){0.f,0.f,0.f,0.f,0.f,0.f,0.f,0.f};
    wmma_tiles_f32<KDIM, NT>(A, lda, W, m0, n0, acc);
#pragma unroll
    for (int t = 0; t < NT; ++t)
        store_tile_f32(C, ldc, m0, n0 + t * 16, acc[t]);
}

// ---------------- dt-proj GEMM with double-bias + softplus ----------------
// Reference: dt = einsum + b ; dt = softplus(dt + b)  => softplus(gemm + 2b)
template<int KDIM, int NT>
__global__ void gemm_wmma_dt_kernel(const float* __restrict__ A, int lda,
                                    const float* __restrict__ W,
                                    const float* __restrict__ bias,
                                    float* __restrict__ C, int ldc) {
    const int m0 = (blockIdx.x * 4 + (threadIdx.x >> 5)) * 16;
    const int n0 = blockIdx.y * (NT * 16);
    v8f acc[NT];
#pragma unroll
    for (int t = 0; t < NT; ++t) acc[t] = (v8f){0.f,0.f,0.f,0.f,0.f,0.f,0.f,0.f};
    wmma_tiles_f32<KDIM, NT>(A, lda, W, m0, n0, acc);

    const int lane = threadIdx.x & 31;
    const int lm   = lane & 15;
    const int rofs = (lane >> 4) << 3;
#pragma unroll
    for (int t = 0; t < NT; ++t) {
        const int col = n0 + t * 16 + lm;
        const float b2 = 2.0f * bias[col];
#pragma unroll
        for (int r = 0; r < 8; ++r) {
            float v = acc[t][r] + b2;
            v = (v > 20.0f) ? v : __logf(1.0f + __expf(v));   // softplus
            C[(size_t)(m0 + r + rofs) * (size_t)ldc + col] = v;
        }
    }
}

// ---------------- Causal depthwise conv (k=4) + bias + SiLU ----------------
// x_fwd lives in XR[:, 0:512] (ld = 1024). Output xs (ld = 512).
__global__ void conv_silu_kernel(const float* __restrict__ XR,
                                 const float* __restrict__ conv_w,
                                 const float* __restrict__ conv_b,
                                 float* __restrict__ xs) {
    int idx = blockIdx.x * blockDim.x + threadIdx.x;
    if (idx >= MROWS * D_INNER) return;
    const int d   = idx & (D_INNER - 1);
    const int row = idx >> 9;                   // /512
    const int l   = row & (L_SEQ - 1);
    const int b   = row >> 11;                  // /2048
    float acc = conv_b[d];
#pragma unroll
    for (int j = 0; j < D_CONV; ++j) {
        const int ls = l - (D_CONV - 1) + j;
        if (ls >= 0)
            acc += XR[((size_t)(b * L_SEQ + ls)) * 1024 + d] * conv_w[d * D_CONV + j];
    }
    const float sig = 1.0f / (1.0f + __expf(-acc));
    xs[idx] = acc * sig;                        // SiLU
}

// ---------------- Selective scan (sequential in L) ----------------
// grid = (D_INNER/128, B_SZ, 2 dirs); block = 128 threads, one channel each.
// h[16] state in registers; per-step B/C (32 floats) broadcast through LDS.
__global__ void scan_kernel(const float* __restrict__ xs,
                            const float* __restrict__ xdbl,   // (MROWS, 48)
                            const float* __restrict__ dt,     // (MROWS, 512)
                            const float* __restrict__ A_log,  // (512, 16)
                            const float* __restrict__ D_param,
                            float* __restrict__ y_fwd,
                            float* __restrict__ y_bwd) {
    const int tid = threadIdx.x;
    const int d   = blockIdx.x * 128 + tid;
    const int b   = blockIdx.y;
    const int dir = blockIdx.z;
    float* __restrict__ yout = dir ? y_bwd : y_fwd;

    float Acoef[D_STATE];
#pragma unroll
    for (int n = 0; n < D_STATE; ++n)
        Acoef[n] = -__expf(A_log[d * D_STATE + n]);   // A = -exp(A_log)
    float h[D_STATE];
#pragma unroll
    for (int n = 0; n < D_STATE; ++n) h[n] = 0.0f;
    const float Dp = D_param[d];

    __shared__ float sBC[32];                   // [0:16) = B, [16:32) = C

    for (int l = 0; l < L_SEQ; ++l) {
        const int tl  = dir ? (L_SEQ - 1 - l) : l;
        const size_t row = (size_t)b * L_SEQ + tl;
        if (tid < 32) sBC[tid] = xdbl[row * 48 + DT_RANK + tid];
        __syncthreads();
        const float dtv = dt[row * D_INNER + d];
        const float xv  = xs[row * D_INNER + d];
        float y = 0.0f;
#pragma unroll
        for (int n = 0; n < D_STATE; ++n) {
            const float dA = __expf(dtv * Acoef[n]);
            h[n] = h[n] * dA + dtv * sBC[n] * xv;       // h*dA + dt*B*x
            y += h[n] * sBC[16 + n];                    // y = <h, C>
        }
        yout[row * D_INNER + d] = y + xv * Dp;
        __syncthreads();
    }
}

// ---------------- Combine: (y_fwd + y_bwd) * silu(res) ----------------
__global__ void combine_kernel(const float* __restrict__ y_fwd,
                               const float* __restrict__ y_bwd,
                               const float* __restrict__ XR,   // res at cols 512..1023
                               float* __restrict__ yc) {
    int idx = blockIdx.x * blockDim.x + threadIdx.x;
    if (idx >= MROWS * D_INNER) return;
    const int d   = idx & (D_INNER - 1);
    const int row = idx >> 9;
    const float r   = XR[(size_t)row * 1024 + D_INNER + d];
    const float sig = 1.0f / (1.0f + __expf(-r));
    yc[idx] = (y_fwd[idx] + y_bwd[idx]) * (r * sig);
}

// ---------------- Host side ----------------
extern "C" void kernel_launch(void* const* d_in, const int* in_sizes, int n_in,
                              void* d_out, int out_size, void* d_ws, size_t ws_size,
                              hipStream_t stream) {
    (void)in_sizes; (void)n_in; (void)out_size; (void)ws_size;
    const float* x         = (const float*)d_in[0];   // (B,L,256)
    const float* in_proj_w = (const float*)d_in[1];   // (1024,256)
    const float* conv_w    = (const float*)d_in[2];   // (512,1,4)
    const float* conv_b    = (const float*)d_in[3];   // (512,)
    const float* x_proj_w  = (const float*)d_in[4];   // (48,512)
    const float* dt_proj_w = (const float*)d_in[5];   // (512,16)
    const float* dt_proj_b = (const float*)d_in[6];   // (512,)
    const float* A_log     = (const float*)d_in[7];   // (512,16)
    const float* D_param   = (const float*)d_in[8];   // (512,)
    const float* out_proj_w= (const float*)d_in[9];   // (256,512)
    float* out = (float*)d_out;

    // Workspace layout (floats)
    float* XR   = (float*)d_ws;                          // MROWS x 1024
    float* xs   = XR   + (size_t)MROWS * 1024;           // MROWS x 512
    float* xdbl = xs   + (size_t)MROWS * 512;            // MROWS x 48
    float* dt   = xdbl + (size_t)MROWS * 48;             // MROWS x 512
    float* yf   = dt   + (size_t)MROWS * 512;            // MROWS x 512
    float* yb   = yf   + (size_t)MROWS * 512;            // MROWS x 512
    float* yc   = xs;                                    // reuse xs after scan

    // 1) in_proj: (8192,256) x (1024,256)^T -> XR (8192,1024)
    gemm_wmma_kernel<DIMV, 4><<<dim3(MROWS / 64, 1024 / 64), 128, 0, stream>>>(
        x, DIMV, in_proj_w, XR, 1024);

    // 2) depthwise causal conv + SiLU -> xs (8192,512)
    conv_silu_kernel<<<(MROWS * D_INNER) / 256, 256, 0, stream>>>(
        XR, conv_w, conv_b, xs);

    // 3) x_proj: (8192,512) x (48,512)^T -> xdbl (8192,48). N=48 -> NT=3.
    gemm_wmma_kernel<D_INNER, 3><<<dim3(MROWS / 64, 1), 128, 0, stream>>>(
        xs, D_INNER, x_proj_w, xdbl, 48);

    // 4) dt_proj: (8192,16 from xdbl ld48) x (512,16)^T + 2b, softplus -> dt
    gemm_wmma_dt_kernel<DT_RANK, 4><<<dim3(MROWS / 64, D_INNER / 64), 128, 0, stream>>>(
        xdbl, 48, dt_proj_w, dt_proj_b, dt, D_INNER);

    // 5) bidirectional selective scan -> yf, yb
    scan_kernel<<<dim3(D_INNER / 128, B_SZ, 2), 128, 0, stream>>>(
        xs, xdbl, dt, A_log, D_param, yf, yb);

    // 6) combine with silu(res) -> yc
    combine_kernel<<<(MROWS * D_INNER) / 256, 256, 0, stream>>>(yf, yb, XR, yc);

    // 7) out_proj: (8192,512) x (256,512)^T -> out (8192,256)
    gemm_wmma_kernel<D_INNER, 4><<<dim3(MROWS / 64, DIMV / 64), 128, 0, stream>>>(
        yc, D_INNER, out_proj_w, out, DIMV);
}